// Simple_window_attention_25598005084366
// MI455X (gfx1250) — compile-verified
//
#include <hip/hip_runtime.h>
#include <hip/hip_bf16.h>

// CDNA5 / gfx1250 — wave32, WMMA 16x16 tiles.
typedef __attribute__((ext_vector_type(2))) float v2f;
typedef __attribute__((ext_vector_type(8))) float v8f;

// Reference collapses (LayerNorm over a size-1 axis == bias, exactly) to:
//   out[j] = ln2_bias[0] * sum_{i=0..2047} W_out[i*256 + j]
// Implemented as ones(1x2048) @ W_out via V_WMMA_F32_16X16X4_F32 (fp32, lossless).
//
// Grid: 16 blocks, one per 16-column tile of the 256-wide output.
// Block: 256 threads = 8 waves; wave w reduces rows [w*256, w*256+256) in
// 64 WMMA K-steps (K=4 each), then a cross-wave LDS reduction finishes.
__global__ __launch_bounds__(256) void win_attn_colsum_wmma(
    const float* __restrict__ W_out,   // (2048, 256) row-major
    const float* __restrict__ ln2_bias,// (1,)
    float* __restrict__ out)           // (256,)
{
    const int tile  = blockIdx.x;        // 0..15 -> columns [16*tile, 16*tile+16)
    const int c0    = tile << 4;
    const int lane  = threadIdx.x & 31;
    const int wave  = threadIdx.x >> 5;  // 0..7
    const int n     = lane & 15;         // N index within tile
    const int khalf = lane >> 4;         // 0: K={0,1}, 1: K={2,3} (32-bit operand layout)

    __shared__ float red[8 * 16];

    // A = 16x4 all-ones: every lane holds {K even, K odd} = {1,1}.
    v2f a; a.x = 1.0f; a.y = 1.0f;
    v8f acc = {};

    // B tile pointer for this wave's K-slice: rows [wave*256 ...), column c0+n.
    // Lane layout for 4x16 fp32 B: lanes 0-15 carry rows k0+0,k0+1 of column n;
    // lanes 16-31 carry rows k0+2,k0+3.
    const float* __restrict__ Wp = W_out + ((wave << 8) + (khalf << 1)) * 256 + c0 + n;

    #pragma unroll 4
    for (int s = 0; s < 64; ++s) {       // 64 steps x K=4 = 256 rows per wave
        v2f b;
        b.x = Wp[0];
        b.y = Wp[256];
        Wp += 4 * 256;
        // D = A(ones) x B + C  -> every row of D accumulates the column sums.
        acc = __builtin_amdgcn_wmma_f32_16x16x4_f32(
            /*neg_a=*/false, a, /*neg_b=*/false, b,
            /*c_mod=*/(short)0, acc, /*reuse_a=*/false, /*reuse_b=*/false);
    }

    // Row M=0 of D lives in VGPR0 on lanes 0..15 (N = lane).
    if (lane < 16) red[(wave << 4) + lane] = acc[0];
    __syncthreads();

    if (threadIdx.x < 16) {
        float s = 0.0f;
        #pragma unroll
        for (int w = 0; w < 8; ++w) s += red[(w << 4) + threadIdx.x];
        out[c0 + threadIdx.x] = ln2_bias[0] * s;
    }
}

extern "C" void kernel_launch(void* const* d_in, const int* in_sizes, int n_in,
                              void* d_out, int out_size, void* d_ws, size_t ws_size,
                              hipStream_t stream) {
    // setup_inputs order:
    // 0:x 1:W_qkv 2:W_out 3:ln1_scale 4:ln1_bias 5:ln2_scale 6:ln2_bias 7:carry
    const float* W_out    = (const float*)d_in[2];
    const float* ln2_bias = (const float*)d_in[6];
    float* out            = (float*)d_out;

    (void)in_sizes; (void)n_in; (void)out_size; (void)d_ws; (void)ws_size;

    win_attn_colsum_wmma<<<16, 256, 0, stream>>>(W_out, ln2_bias, out);
}